// AttentionLayer_83760452206702
// MI455X (gfx1250) — compile-verified
//
#include <hip/hip_runtime.h>
#include <hip/hip_bf16.h>

typedef __attribute__((ext_vector_type(2))) float v2f;
typedef __attribute__((ext_vector_type(8))) float v8f;

#define B_DIM 4
#define TQ 512
#define TV 512
#define DM 256
#define UN 128
#define TPB_T 8   // query rows per block in the score kernel

static __device__ __forceinline__ float fast_tanh(float x) {
#if __has_builtin(__builtin_amdgcn_tanhf)
    return __builtin_amdgcn_tanhf(x);   // V_TANH_F32 (CDNA5 trans op)
#else
    return tanhf(x);
#endif
}

// One wave computes one 16x16 fp32 C tile with V_WMMA_F32_16X16X4_F32,
// looping K in steps of 4. Row-major A [lda], B [ldb], C [ldc].
// ISA 7.12.2 layout: lane L -> m/n = L%16, half = L/16.
//   A VGPRs: a[v] = A[m][k0 + 2*half + v]
//   B VGPRs: b[v] = B[k0 + 2*half + v][n]
//   C VGPRs: c[r] = C[r + 8*half][n]
static __device__ __forceinline__ void wmma_tile_f32(
    const float* __restrict__ A, int lda,
    const float* __restrict__ B, int ldb,
    float* __restrict__ C, int ldc,
    int m0, int n0, int K)
{
    const int lane = threadIdx.x & 31;
    const int half = lane >> 4;     // 0 or 1
    const int mn   = lane & 15;     // row of A-tile / col of B-tile
    v8f acc = {0.f, 0.f, 0.f, 0.f, 0.f, 0.f, 0.f, 0.f};
    const float* __restrict__ arow = A + (size_t)(m0 + mn) * lda + 2 * half;
    const float* __restrict__ bcol = B + (size_t)(2 * half) * ldb + (n0 + mn);
#pragma unroll 2
    for (int k0 = 0; k0 < K; k0 += 4) {
        v2f a, b;
        a.x = arow[k0 + 0];
        a.y = arow[k0 + 1];
        b.x = bcol[(size_t)(k0 + 0) * ldb];
        b.y = bcol[(size_t)(k0 + 1) * ldb];
        // 8 args: (neg_a, A, neg_b, B, c_mod, C, reuse_a, reuse_b)
        acc = __builtin_amdgcn_wmma_f32_16x16x4_f32(
            false, a, false, b, (short)0, acc, false, false);
    }
#pragma unroll
    for (int r = 0; r < 8; ++r) {
        C[(size_t)(m0 + r + 8 * half) * ldc + (n0 + mn)] = acc[r];
    }
}

// Phase 1: Q = query*W1 [2048x128], K = value*W2 [2048x128]; fp32 WMMA GEMM.
// 2048 tiles total (1024 per projection), 1 wave per 16x16 tile.
__global__ __launch_bounds__(256) void proj_kernel(
    const float* __restrict__ query, const float* __restrict__ value,
    const float* __restrict__ W1, const float* __restrict__ W2,
    float* __restrict__ Qo, float* __restrict__ Ko)
{
    const int wave  = (int)((blockIdx.x * blockDim.x + threadIdx.x) >> 5);
    const int which = wave >> 10;          // 0 = Q projection, 1 = K projection
    const int tile  = wave & 1023;
    const int mt    = tile >> 3;           // 0..127  (2048/16 M-tiles)
    const int nt    = tile & 7;            // 0..7    (128/16  N-tiles)
    const float* A  = which ? value : query;
    const float* Bm = which ? W2 : W1;
    float* C        = which ? Ko : Qo;
    wmma_tile_f32(A, DM, Bm, UN, C, UN, mt * 16, nt * 16, DM);
}

// Phase 2: scores + softmax. One block of 512 threads handles (b, 8 query rows).
// Thread s keeps its K row (128 f32) in VGPRs, loaded once with b128 loads;
// the 8 q rows are broadcast from LDS. Inner loop: ds_load + v_tanh_f32 + fma.
// Per t: block-wide softmax over the 512 scores, write attn row (coalesced).
__global__ __launch_bounds__(512) void score_softmax_kernel(
    const float* __restrict__ Q, const float* __restrict__ Kmat,
    const float* __restrict__ scale, float* __restrict__ attn)
{
    const int b    = (int)blockIdx.x >> 6;   // 64 blocks per batch (512/8)
    const int t0   = ((int)blockIdx.x & 63) * TPB_T;
    const int tid  = (int)threadIdx.x;       // s index, 0..511
    __shared__ float q_sh[TPB_T][UN];
    __shared__ float sc_sh[UN];
    __shared__ float red[16];

    if (tid < UN) sc_sh[tid] = scale[tid];
    {
        const int tt = tid >> 7;             // 0..3
        const int uu = tid & 127;
        q_sh[tt][uu]     = Q[(size_t)(b * TQ + t0 + tt) * UN + uu];
        q_sh[tt + 4][uu] = Q[(size_t)(b * TQ + t0 + tt + 4) * UN + uu];
    }

    // This thread's K row -> registers (32 x global_load_b128)
    float kreg[UN];
    {
        const float4* __restrict__ krow4 =
            (const float4*)(Kmat + ((size_t)(b * TV + tid)) * UN);
#pragma unroll
        for (int i = 0; i < UN / 4; ++i) {
            const float4 v = krow4[i];
            kreg[4 * i + 0] = v.x;
            kreg[4 * i + 1] = v.y;
            kreg[4 * i + 2] = v.z;
            kreg[4 * i + 3] = v.w;
        }
    }
    __syncthreads();

    const int w = tid >> 5;
    for (int tt = 0; tt < TPB_T; ++tt) {
        float s = 0.f;
#pragma unroll
        for (int u = 0; u < UN; ++u) {
            s = __builtin_fmaf(sc_sh[u], fast_tanh(q_sh[tt][u] + kreg[u]), s);
        }

        // block max (wave32 shuffle, then LDS across 16 waves)
        float m = s;
#pragma unroll
        for (int off = 16; off > 0; off >>= 1)
            m = fmaxf(m, __shfl_xor(m, off, 32));
        if ((tid & 31) == 0) red[w] = m;
        __syncthreads();
        float bmax = red[0];
#pragma unroll
        for (int i = 1; i < 16; ++i) bmax = fmaxf(bmax, red[i]);

        const float e = __expf(s - bmax);

        __syncthreads();                     // all reads of red[] done
        float sum = e;
#pragma unroll
        for (int off = 16; off > 0; off >>= 1)
            sum += __shfl_xor(sum, off, 32);
        if ((tid & 31) == 0) red[w] = sum;
        __syncthreads();
        float bsum = 0.f;
#pragma unroll
        for (int i = 0; i < 16; ++i) bsum += red[i];

        attn[(size_t)(b * TQ + t0 + tt) * TV + tid] = e / bsum;
        __syncthreads();                     // red[] reused next iteration
    }
}

// Phase 3: ctx[b] = attn[b] (512x512) * value[b] (512x256); fp32 WMMA GEMM.
// 4 * 32 * 16 = 2048 tiles, 1 wave per tile.
__global__ __launch_bounds__(256) void context_kernel(
    const float* __restrict__ attn, const float* __restrict__ value,
    float* __restrict__ ctx)
{
    const int wave = (int)((blockIdx.x * blockDim.x + threadIdx.x) >> 5);
    const int b    = wave >> 9;
    const int t    = wave & 511;
    const int mt   = t >> 4;               // 0..31
    const int nt   = t & 15;               // 0..15
    const float* A  = attn  + (size_t)b * TQ * TV;
    const float* Bv = value + (size_t)b * TV * DM;
    float* C        = ctx   + (size_t)b * TQ * DM;
    wmma_tile_f32(A, TV, Bv, DM, C, DM, mt * 16, nt * 16, TV);
}

extern "C" void kernel_launch(void* const* d_in, const int* in_sizes, int n_in,
                              void* d_out, int out_size, void* d_ws, size_t ws_size,
                              hipStream_t stream) {
    const float* query = (const float*)d_in[0];   // [4,512,256]
    const float* value = (const float*)d_in[1];   // [4,512,256]
    const float* W1    = (const float*)d_in[2];   // [256,128]
    const float* W2    = (const float*)d_in[3];   // [256,128]
    const float* scale = (const float*)d_in[4];   // [128]

    float* out  = (float*)d_out;
    float* ctx  = out;                              // [4,512,256] = 524288
    float* attn = out + (size_t)B_DIM * TQ * DM;    // [4,512,512] = 1048576

    float* Qw = (float*)d_ws;                       // [2048,128]
    float* Kw = Qw + (size_t)(B_DIM * TQ) * UN;     // [2048,128]

    // 2048 waves, 8 waves/block
    proj_kernel<<<256, 256, 0, stream>>>(query, value, W1, W2, Qw, Kw);
    // 256 blocks: (b, group of 8 t rows); 512 threads = one s each
    score_softmax_kernel<<<(B_DIM * TQ) / TPB_T, TV, 0, stream>>>(Qw, Kw, scale, attn);
    // 2048 waves, 8 waves/block
    context_kernel<<<256, 256, 0, stream>>>(attn, value, ctx);
}